// UniversalEnergyField_77257871721081
// MI455X (gfx1250) — compile-verified
//
#include <hip/hip_runtime.h>
#include <hip/hip_bf16.h>
#include <stdint.h>

#define B_ROWS 8192
#define DIM    2048
#define NST    14
#define TPB    256
#define WAVES  (TPB / 32)

typedef float v2f __attribute__((ext_vector_type(2)));
typedef float v8f __attribute__((ext_vector_type(8)));

#if __has_builtin(__builtin_amdgcn_global_load_async_to_lds_b128)
#define ASYNC_LDS 1
// Parameter types per hipcc diagnostic: pointee is int __attribute__((vector_size(16)))
typedef int v4i_vs __attribute__((vector_size(16)));
typedef __attribute__((address_space(1))) v4i_vs GV4;   // global (prints as __device__)
typedef __attribute__((address_space(3))) v4i_vs LV4;   // LDS    (prints as __shared__)
#endif

// Raw hardware approximations: single v_rcp_f32 / v_rsq_f32 (≈1 ulp), instead
// of the correctly-rounded div/rsqrt Newton chains (~10 VALU ops each).
__device__ __forceinline__ float hrcp(float v)  { return __builtin_amdgcn_rcpf(v); }
__device__ __forceinline__ float hrsq(float v)  { return __builtin_amdgcn_rsqf(v); }

// Branchless fast atan2 (Hastings poly, ~1e-5 abs err) — the hot transcendental
// (~2.5 calls/element).
__device__ __forceinline__ float fatan2f(float y, float x) {
  float ax = __builtin_fabsf(x), ay = __builtin_fabsf(y);
  float mx = fmaxf(ax, ay), mn = fminf(ax, ay);
  float z  = mn * hrcp(fmaxf(mx, 1.0e-37f));
  float z2 = z * z;
  float p  = fmaf(z2, 0.0208351f, -0.0851330f);
  p = fmaf(z2, p, 0.1801410f);
  p = fmaf(z2, p, -0.3302995f);
  p = fmaf(z2, p, 0.9998660f);
  float a = p * z;
  a = (ay > ax) ? (1.5707963705f - a) : a;
  a = (x < 0.0f) ? (3.1415927410f - a) : a;
  return __builtin_copysignf(a, y);
}

__device__ __forceinline__ float sigmoidf(float v) {
  return hrcp(1.0f + __expf(-v));
}

__device__ __forceinline__ float wave_sum(float v) {
  v += __shfl_down(v, 16, 32);
  v += __shfl_down(v, 8, 32);
  v += __shfl_down(v, 4, 32);
  v += __shfl_down(v, 2, 32);
  v += __shfl_down(v, 1, 32);
  return v;
}

// coeff[i] = 1 + sum_{j>i} coupling[i][j] - sum_{j<i} coupling[j][i]
__global__ void uef_coeff(const float* __restrict__ coupling, float* __restrict__ coeff) {
  int i = threadIdx.x;
  if (i < NST) {
    float s = 1.0f;
    for (int j = i + 1; j < NST; ++j) s += coupling[i * NST + j];
    for (int j = 0; j < i; ++j)       s -= coupling[j * NST + i];
    coeff[i] = s;
  }
}

__global__ __launch_bounds__(TPB) void uef_main(
    const float* __restrict__ x,
    const float* __restrict__ sw,
    const float* __restrict__ alphas,
    const float* __restrict__ res_w,
    const float* __restrict__ coeff,
    float* __restrict__ out)
{
  __shared__ __align__(16) float lx[DIM];   // this row
  __shared__ __align__(16) float lp[DIM];   // previous row (quantum imag part)
  __shared__ float red[WAVES * 10];
  __shared__ float fin[10];
  __shared__ float cns[4];

  const int t    = threadIdx.x;
  const int row  = blockIdx.x;
  const int prow = (row + B_ROWS - 1) & (B_ROWS - 1);
  const float* xr = x + (size_t)row  * DIM;
  const float* pr = x + (size_t)prow * DIM;

  // ---- stage both rows into LDS (CDNA5 async global->LDS path) ----
#if ASYNC_LDS
  {
    __builtin_amdgcn_global_load_async_to_lds_b128((GV4*)(xr + 4 * t),        (LV4*)(lx + 4 * t),        0, 0);
    __builtin_amdgcn_global_load_async_to_lds_b128((GV4*)(xr + 1024 + 4 * t), (LV4*)(lx + 1024 + 4 * t), 0, 0);
    __builtin_amdgcn_global_load_async_to_lds_b128((GV4*)(pr + 4 * t),        (LV4*)(lp + 4 * t),        0, 0);
    __builtin_amdgcn_global_load_async_to_lds_b128((GV4*)(pr + 1024 + 4 * t), (LV4*)(lp + 1024 + 4 * t), 0, 0);
#if __has_builtin(__builtin_amdgcn_s_wait_asynccnt)
    __builtin_amdgcn_s_wait_asynccnt(0);
#else
    asm volatile("s_wait_asynccnt 0" ::: "memory");
#endif
  }
#else
  {
    float4 a = ((const float4*)xr)[t];
    float4 b = ((const float4*)xr)[256 + t];
    float4 c = ((const float4*)pr)[t];
    float4 d = ((const float4*)pr)[256 + t];
    ((float4*)lx)[t] = a;  ((float4*)lx)[256 + t] = b;
    ((float4*)lp)[t] = c;  ((float4*)lp)[256 + t] = d;
  }
#endif
  __syncthreads();

  // ---- phase 2: per-thread partial statistics over elements [8t, 8t+8) ----
  const int d0 = t * 8;
  float xs[8], fv[8];
#pragma unroll
  for (int k = 0; k < 8; ++k) xs[k] = lx[d0 + k];

  float s1 = 0, s2 = 0, ag = 0, aq = 0, te1 = 0, te2 = 0, at = 0, f1 = 0, f2 = 0, af = 0;
#pragma unroll
  for (int k = 0; k < 8; ++k) { s1 += xs[k]; s2 = fmaf(xs[k], xs[k], s2); }
#pragma unroll
  for (int k = 0; k < 8; k += 2) ag += fatan2f(xs[k + 1], xs[k]);
#pragma unroll
  for (int k = 0; k < 8; ++k) aq += fatan2f(lp[d0 + k], xs[k]);

  // toroidal: cos/sin via pair normalization (no trig needed)
#pragma unroll
  for (int g = 0; g < 2; ++g) {
    float a0 = xs[4 * g], b0 = xs[4 * g + 1], a1 = xs[4 * g + 2], b1 = xs[4 * g + 3];
    float r0 = hrsq(fmaf(a0, a0, b0 * b0));
    float r1 = hrsq(fmaf(a1, a1, b1 * b1));
    float ca = a0 * r0, sa = b0 * r0, cb = a1 * r1, sb = b1 * r1;
    te1 += ca + sa + cb + sb;
    te2 += ca * ca + sa * sa + cb * cb + sb * sb;
    at  += fatan2f(cb, ca) + fatan2f(sb, sa);
  }

  // fractal: 4-tap circular FIR with phi^(-k/4) taps, rolling window
  {
    float w1 = lx[(d0 - 1) & (DIM - 1)];
    float w2 = lx[(d0 - 2) & (DIM - 1)];
    float w3 = lx[(d0 - 3) & (DIM - 1)];
#pragma unroll
    for (int k = 0; k < 8; ++k) {
      float cur = xs[k];
      float f = fmaf(0.69704196f, w3, fmaf(0.78615135f, w2, fmaf(0.88665175f, w1, cur)));
      fv[k] = f; f1 += f; f2 = fmaf(f, f, f2);
      w3 = w2; w2 = w1; w1 = cur;
    }
  }
#pragma unroll
  for (int k = 0; k < 8; k += 2) af += fatan2f(fv[k + 1], fv[k]);

  // ---- deterministic block reduction: shuffle within wave, LDS across waves ----
  const int wv = t >> 5, lane = t & 31;
  {
    float vals[10] = {s1, s2, ag, aq, te1, te2, at, f1, f2, af};
#pragma unroll
    for (int q = 0; q < 10; ++q) {
      float r = wave_sum(vals[q]);
      if (lane == 0) red[wv * 10 + q] = r;
    }
  }
  __syncthreads();
  if (t < 10) {
    float s = 0.0f;
#pragma unroll
    for (int w = 0; w < WAVES; ++w) s += red[w * 10 + t];
    fin[t] = s;
  }
  __syncthreads();

  // ---- epilogue in wave 0: scalar stats + WMMA einsum + preservation chain ----
  if (wv == 0) {
    const float invD = 1.0f / (float)DIM;
    const float inv_sqrt_d = 0.022097086912079608f;   // 1/sqrt(2048)
    const float inv_dm1 = 1.0f / (float)(DIM - 1);

    float s1t = fin[0], s2t = fin[1];
    float cg = sqrtf(s2t) * inv_sqrt_d;
    float fg = sqrtf(fmaxf(0.f, s2t - s1t * s1t * invD) * inv_dm1);
    float th_g = fin[2] * (1.0f / 1024.0f);
    float th_q = fin[3] * (1.0f / 2048.0f);
    float e1 = fin[4], e2 = fin[5];
    float ct = sqrtf(e2) * inv_sqrt_d;
    float ft = sqrtf(fmaxf(0.f, e2 - e1 * e1 * invD) * inv_dm1);
    float th_t = fin[6] * (1.0f / 1024.0f);
    float g1 = fin[7], g2 = fin[8];
    float cf = sqrtf(g2) * inv_sqrt_d;
    float ff = sqrtf(fmaxf(0.f, g2 - g1 * g1 * invD) * inv_dm1);
    float th_f = fin[9] * (1.0f / 1024.0f);

    float hT[4] = {th_g, th_q, th_t, th_f};
    float hC[4] = {cg, cg, ct, cf};
    float hF[4] = {fg, fg, ft, ff};

    float sigs[4] = {0.f, 0.f, 0.f, 0.f};

#if __has_builtin(__builtin_amdgcn_wmma_f32_16x16x4_f32)
    {
      // sel[v][r] = sum_k h[v][k] * res_w[r, r%3, k]  as a 16x16x4 WMMA:
      // A (16x4): rows 0..3 = (theta, coh, freq, 0); lanes 0-15 hold K=0,1 in
      // v[0],v[1]; lanes 16-31 hold K=2,3.  B (4x16): cols 0..5 = selected res_w.
      int m = lane & 15;
      v2f A; A[0] = 0.f; A[1] = 0.f;
      if (m < 4) {
        if (lane < 16) { A[0] = hT[m]; A[1] = hC[m]; }
        else           { A[0] = hF[m]; A[1] = 0.f;  }
      }
      v2f Bm; Bm[0] = 0.f; Bm[1] = 0.f;
      if (m < 6) {
        const float* wb = res_w + m * 27 + (m % 3) * 3;
        if (lane < 16) { Bm[0] = wb[0]; Bm[1] = wb[1]; }
        else           { Bm[0] = wb[2]; Bm[1] = 0.f;  }
      }
      v8f Cz = {0.f, 0.f, 0.f, 0.f, 0.f, 0.f, 0.f, 0.f};
      v8f Dm = __builtin_amdgcn_wmma_f32_16x16x4_f32(
          false, A, false, Bm, (short)0, Cz, false, false);
      // D: lane n (0..5) holds sel[v][n] in Dm[v], v=0..3
      if (lane < 6) {
#pragma unroll
        for (int v = 0; v < 4; ++v) sigs[v] = sigmoidf(Dm[v]);
      }
#pragma unroll
      for (int off = 4; off >= 1; off >>= 1) {
#pragma unroll
        for (int v = 0; v < 4; ++v) sigs[v] += __shfl_down(sigs[v], off, 32);
      }
    }
#else
    if (lane == 0) {
#pragma unroll
      for (int v = 0; v < 4; ++v) {
        float s = 0.f;
        for (int r = 0; r < 6; ++r) {
          const float* wb = res_w + r * 27 + (r % 3) * 3;
          s += sigmoidf(hT[v] * wb[0] + hC[v] * wb[1] + hF[v] * wb[2]);
        }
        sigs[v] = s;
      }
    }
#endif

    if (lane == 0) {
      const float THG = 2.3998277f;  // 137.5 deg
      float P[4];
#pragma unroll
      for (int v = 0; v < 4; ++v) {
        float boost = fmaf(0.1f, sigs[v], 1.0f);
        float ta = __expf(-__builtin_fabsf(hT[v] - THG));
        float dd = hC[v] - 0.618f;
        float cal = __expf(-10.0f * dd * dd);
        P[v] = ta * cal * boost;
      }
      const float* swr = sw + (size_t)row * NST;
      float cgen = 0.f, Cq = 0.f, Ct = 0.f, Cf = 0.f;
#pragma unroll
      for (int i = 0; i < NST; ++i) {
        float wi = alphas[i] * coeff[i] * swr[i];
        if (i == 2)       Cq = P[1] * wi;
        else if (i == 9)  Ct = P[2] * wi;
        else if (i == 10) Cf = P[3] * wi;
        else              cgen += wi;
      }
      cns[0] = P[0] * cgen + Cq;  // coefficient on x (real)
      cns[1] = Cq;                // coefficient on prev row (imag)
      cns[2] = Ct;                // toroidal energy coefficient
      cns[3] = Cf;                // fractal energy coefficient
    }
  }
  __syncthreads();

  // ---- phase 3: fused output  out = sqrt(real^2 + imag^2) ----
  const float Cx = cns[0], Cq = cns[1], Ct = cns[2], Cf = cns[3];
  float res[8];
#pragma unroll
  for (int k = 0; k < 8; ++k) {
    int d = d0 + k;
    int j = d & 1023;
    float a = lx[2 * j], b = lx[2 * j + 1];
    float r = hrsq(fmaf(a, a, b * b));
    float et = ((d < 1024) ? a : b) * r;           // cos/sin half of toroidal energy
    float re = fmaf(Cx, xs[k], fmaf(Ct, et, Cf * fv[k]));
    float im = Cq * lp[d];
    res[k] = sqrtf(fmaf(re, re, im * im));
  }
  float4* o = (float4*)(out + (size_t)row * DIM + d0);
  o[0] = make_float4(res[0], res[1], res[2], res[3]);
  o[1] = make_float4(res[4], res[5], res[6], res[7]);
}

extern "C" void kernel_launch(void* const* d_in, const int* in_sizes, int n_in,
                              void* d_out, int out_size, void* d_ws, size_t ws_size,
                              hipStream_t stream) {
  const float* x        = (const float*)d_in[0];
  const float* sweights = (const float*)d_in[1];
  const float* alphas   = (const float*)d_in[2];
  const float* coupling = (const float*)d_in[3];
  const float* res_w    = (const float*)d_in[4];
  float* out   = (float*)d_out;
  float* coeff = (float*)d_ws;   // 14 floats of scratch

  uef_coeff<<<1, 32, 0, stream>>>(coupling, coeff);
  uef_main<<<B_ROWS, TPB, 0, stream>>>(x, sweights, alphas, res_w, coeff, out);
}